// GCN_53678501266191
// MI455X (gfx1250) — compile-verified
//
#include <hip/hip_runtime.h>

// ---------------- problem constants (match reference) ----------------
#define N_NODES  50000     // 3125 * 16  -> exact 16-row WMMA tiling
#define N_EDGES  600000
#define N_GRAPHS 256
#define IN_C     64
#define EMB      128
#define HID      128
#define OUTC     36

typedef float v2f __attribute__((ext_vector_type(2)));
typedef float v8f __attribute__((ext_vector_type(8)));

// ---------------- small utility kernels ----------------
__global__ __launch_bounds__(256) void fill_kernel(float* p, float v, int n) {
    int i = blockIdx.x * 256 + threadIdx.x;
    if (i < n) p[i] = v;
}

// out[i][c] = bias[c]  (scatter target starts at the bias; segment_sum adds on top)
__global__ __launch_bounds__(256) void init_bias_kernel(float* __restrict__ h,
                                                        const float* __restrict__ b) {
    int i = blockIdx.x * 256 + threadIdx.x;
    if (i < N_NODES * EMB) h[i] = b[i & (EMB - 1)];
}

__global__ __launch_bounds__(256) void relu_kernel(float* h, int n) {
    int i = blockIdx.x * 256 + threadIdx.x;
    if (i < n) h[i] = fmaxf(h[i], 0.0f);
}

// ---------------- degree / normalization ----------------
// deg[col] += w  over real edges, +1 for the self loop of every node
__global__ __launch_bounds__(256) void degree_kernel(const int* __restrict__ col,
                                                     const float* __restrict__ w,
                                                     float* __restrict__ deg) {
    int e = blockIdx.x * 256 + threadIdx.x;
    if (e < N_EDGES)                  atomicAdd(&deg[col[e]], w[e]);
    else if (e < N_EDGES + N_NODES)   atomicAdd(&deg[e - N_EDGES], 1.0f);
}

// deg -> dis = deg>0 ? rsqrt(max(deg,1e-30)) : 0   (in place)
__global__ __launch_bounds__(256) void dis_kernel(float* deg) {
    int i = blockIdx.x * 256 + threadIdx.x;
    if (i < N_NODES) {
        float d = deg[i];
        deg[i] = (d > 0.0f) ? rsqrtf(fmaxf(d, 1e-30f)) : 0.0f;
    }
}

// ---------------- dense GEMM via V_WMMA_F32_16X16X4_F32 ----------------
// Y[N_NODES x 128] = X[N_NODES x K] @ W[K x 128].
// One block = 16 output rows; wave w (of 8) owns 16 output cols.
// Fragment layouts per CDNA5 ISA 7.12.2:
//   A (16x4 f32, 2 VGPR): lanes0-15 M=lane K={0,1}; lanes16-31 M=lane-16 K={2,3}
//   B (4x16 f32, 2 VGPR): lane&15 = N; half-wave selects K pair (mirror of A)
//   C/D (16x16 f32, 8 VGPR): VGPR r -> M = r + 8*(lane>>4), N = lane&15
template <int K>
__global__ __launch_bounds__(256) void gemm_wmma_kernel(const float* __restrict__ X,
                                                        const float* __restrict__ W,
                                                        float* __restrict__ Y) {
    const int wave = threadIdx.x >> 5;          // 0..7 -> 16-col tile
    const int lane = threadIdx.x & 31;
    const int g    = lane >> 4;                 // half-wave group (0/1)
    const int m    = lane & 15;                 // row for A, col for B/C/D
    const int nb   = wave * 16;
    const long mT  = (long)blockIdx.x * 16;

    const float* xrow = X + (mT + m) * (long)K;
    v8f acc = {};
#pragma unroll
    for (int k0 = 0; k0 < K; k0 += 4) {
        const int k = k0 + g * 2;
        v2f a, b;
        a.x = xrow[k];
        a.y = xrow[k + 1];
        b.x = W[(size_t)k * EMB + nb + m];
        b.y = W[(size_t)(k + 1) * EMB + nb + m];
        // (neg_a, A, neg_b, B, c_mod, C, reuse_a, reuse_b)
        acc = __builtin_amdgcn_wmma_f32_16x16x4_f32(false, a, false, b,
                                                    (short)0, acc, false, false);
    }
#pragma unroll
    for (int r = 0; r < 8; ++r)
        Y[(mT + r + g * 8) * EMB + nb + m] = acc[r];
}

// ---------------- edge scatter: out[col] += dis[row]*w*dis[col] * lin[row] ----------------
// One wave32 per edge; lane handles 4 channels (float4) of the 128-wide row.
__global__ __launch_bounds__(256) void scatter_kernel(const float* __restrict__ lin,
                                                      const float* __restrict__ dis,
                                                      const int* __restrict__ row,
                                                      const int* __restrict__ col,
                                                      const float* __restrict__ ew,
                                                      float* __restrict__ out) {
    int e    = blockIdx.x * 8 + (threadIdx.x >> 5);
    int lane = threadIdx.x & 31;
    if (e >= N_EDGES + N_NODES) return;
    int r, c; float w;
    if (e < N_EDGES) { r = row[e]; c = col[e]; w = ew[e]; }
    else             { r = c = e - N_EDGES;    w = 1.0f;  }
    float norm = dis[r] * w * dis[c];
    float4 v = ((const float4*)(lin + (size_t)r * EMB))[lane];
    float* dst = out + (size_t)c * EMB + lane * 4;
    atomicAdd(dst + 0, norm * v.x);
    atomicAdd(dst + 1, norm * v.y);
    atomicAdd(dst + 2, norm * v.z);
    atomicAdd(dst + 3, norm * v.w);
}

// ---------------- segment max / mean pooling ----------------
// h >= 0 after ReLU, so uint bit pattern is monotone -> atomicMax on uint, init 0.
__global__ __launch_bounds__(256) void pool_kernel(const float* __restrict__ h,
                                                   const int* __restrict__ batch,
                                                   float* __restrict__ gmp,
                                                   float* __restrict__ gap,
                                                   float* __restrict__ cnt) {
    int idx = blockIdx.x * 256 + threadIdx.x;
    if (idx >= N_NODES * EMB) return;
    int i = idx >> 7, c = idx & 127;
    int g = batch[i];
    float v = h[idx];
    atomicMax((unsigned int*)&gmp[g * EMB + c], __float_as_uint(v));
    atomicAdd(&gap[g * EMB + c], v);
    if (c == 0) atomicAdd(&cnt[g], 1.0f);
}

// hg = [gmp | gap/cnt | rho]  -> [G, 2*EMB+1]
__global__ __launch_bounds__(256) void build_hg_kernel(const float* __restrict__ gmp,
                                                       const float* __restrict__ gap,
                                                       const float* __restrict__ cnt,
                                                       const float* __restrict__ rho,
                                                       float* __restrict__ hg) {
    const int F = 2 * EMB + 1;
    int idx = blockIdx.x * 256 + threadIdx.x;
    if (idx >= N_GRAPHS * F) return;
    int g = idx / F, j = idx % F;
    float v;
    if (j < EMB)          v = gmp[g * EMB + j];
    else if (j < 2 * EMB) v = gap[g * EMB + (j - EMB)] / cnt[g];
    else                  v = rho[g];
    hg[idx] = v;
}

// ---------------- tiny MLP head (256 rows -> plain VALU is optimal) ----------------
__global__ __launch_bounds__(256) void dense_kernel(const float* __restrict__ in,
                                                    const float* __restrict__ w,
                                                    const float* __restrict__ b,
                                                    float* __restrict__ out,
                                                    int rows, int K, int Ncols, int do_relu) {
    int idx = blockIdx.x * 256 + threadIdx.x;
    if (idx >= rows * Ncols) return;
    int r = idx / Ncols, j = idx % Ncols;
    float acc = b[j];
    const float* inr = in + (size_t)r * K;
    for (int k = 0; k < K; ++k)
        acc = fmaf(inr[k], w[(size_t)k * Ncols + j], acc);
    out[idx] = do_relu ? fmaxf(acc, 0.0f) : acc;
}

// ---------------- launcher ----------------
extern "C" void kernel_launch(void* const* d_in, const int* in_sizes, int n_in,
                              void* d_out, int out_size, void* d_ws, size_t ws_size,
                              hipStream_t stream) {
    const float* x         = (const float*)d_in[0];
    const float* edge_attr = (const float*)d_in[1];
    const float* rho       = (const float*)d_in[2];
    const float* conv0_w   = (const float*)d_in[3];
    const float* conv0_b   = (const float*)d_in[4];
    const float* conv1_w   = (const float*)d_in[5];
    const float* conv1_b   = (const float*)d_in[6];
    const float* mlp0_w    = (const float*)d_in[7];
    const float* mlp0_b    = (const float*)d_in[8];
    const float* mlp1_w    = (const float*)d_in[9];
    const float* mlp1_b    = (const float*)d_in[10];
    const float* out_w     = (const float*)d_in[11];
    const float* out_b     = (const float*)d_in[12];
    const int*   eidx      = (const int*)d_in[13];   // [2, E] (harness int cast)
    const int*   batch     = (const int*)d_in[14];   // [N]
    const int*   row = eidx;
    const int*   col = eidx + N_EDGES;
    float* outp = (float*)d_out;

    // workspace layout (floats); total ~13.05M floats ~= 52.2 MB
    float* ws   = (float*)d_ws;
    float* deg  = ws;                                   // N (becomes dis in place)
    float* bufA = ws + 50176;                           // N*EMB (16B-aligned offset)
    float* bufB = bufA + (size_t)N_NODES * EMB;         // N*EMB
    float* gmp  = bufB + (size_t)N_NODES * EMB;         // G*EMB
    float* gap  = gmp + N_GRAPHS * EMB;                 // G*EMB
    float* cnt  = gap + N_GRAPHS * EMB;                 // G
    float* hg   = cnt + N_GRAPHS;                       // G*(2*EMB+1)
    float* t1   = hg + N_GRAPHS * (2 * EMB + 1);        // G*HID
    float* t2   = t1 + N_GRAPHS * HID;                  // G*HID

    auto cdiv = [](int a, int b) { return (a + b - 1) / b; };
    const int NE = N_EDGES + N_NODES;

    // normalization: deg -> dis
    fill_kernel<<<cdiv(N_NODES, 256), 256, 0, stream>>>(deg, 0.0f, N_NODES);
    degree_kernel<<<cdiv(NE, 256), 256, 0, stream>>>(col, edge_attr, deg);
    dis_kernel<<<cdiv(N_NODES, 256), 256, 0, stream>>>(deg);

    // layer 0: lin0 = x @ W0 (WMMA fp32); out = b0 + segment_sum(norm * lin0[row]); ReLU
    gemm_wmma_kernel<IN_C><<<N_NODES / 16, 256, 0, stream>>>(x, conv0_w, bufA);
    init_bias_kernel<<<cdiv(N_NODES * EMB, 256), 256, 0, stream>>>(bufB, conv0_b);
    scatter_kernel<<<cdiv(NE, 8), 256, 0, stream>>>(bufA, deg, row, col, edge_attr, bufB);
    relu_kernel<<<cdiv(N_NODES * EMB, 256), 256, 0, stream>>>(bufB, N_NODES * EMB);

    // layer 1 (ping-pong the two big buffers)
    gemm_wmma_kernel<EMB><<<N_NODES / 16, 256, 0, stream>>>(bufB, conv1_w, bufA);
    init_bias_kernel<<<cdiv(N_NODES * EMB, 256), 256, 0, stream>>>(bufB, conv1_b);
    scatter_kernel<<<cdiv(NE, 8), 256, 0, stream>>>(bufA, deg, row, col, edge_attr, bufB);
    relu_kernel<<<cdiv(N_NODES * EMB, 256), 256, 0, stream>>>(bufB, N_NODES * EMB);

    // pooling (gmp init 0 is exact: ReLU output >= 0, every graph nonempty)
    fill_kernel<<<cdiv(2 * N_GRAPHS * EMB + N_GRAPHS, 256), 256, 0, stream>>>(
        gmp, 0.0f, 2 * N_GRAPHS * EMB + N_GRAPHS);
    pool_kernel<<<cdiv(N_NODES * EMB, 256), 256, 0, stream>>>(bufB, batch, gmp, gap, cnt);
    build_hg_kernel<<<cdiv(N_GRAPHS * (2 * EMB + 1), 256), 256, 0, stream>>>(
        gmp, gap, cnt, rho, hg);

    // MLP head
    dense_kernel<<<cdiv(N_GRAPHS * HID, 256), 256, 0, stream>>>(
        hg, mlp0_w, mlp0_b, t1, N_GRAPHS, 2 * EMB + 1, HID, 1);
    dense_kernel<<<cdiv(N_GRAPHS * HID, 256), 256, 0, stream>>>(
        t1, mlp1_w, mlp1_b, t2, N_GRAPHS, HID, HID, 1);
    dense_kernel<<<cdiv(N_GRAPHS * OUTC, 256), 256, 0, stream>>>(
        t2, out_w, out_b, outp, N_GRAPHS, HID, OUTC, 0);
}